// ExtendedMoCHILoss_82008105550420
// MI455X (gfx1250) — compile-verified
//
#include <hip/hip_runtime.h>
#include <hip/hip_bf16.h>

// ---------------------------------------------------------------------------
// ExtendedMoCHILoss on gfx1250 (MI455X).
// Bandwidth-bound GEMV vs. anchor, done with V_WMMA_F32_16X16X4_F32:
//   - B matrix (4x16 per step) = 16 rows of h/p, K-chunk of 4
//   - A matrix (16x4) = normalized anchor chunk replicated across all 16 rows
//   => every D-matrix row is identical; c[0] on lane n == dot(row[n&15], a_n)
// Row norms accumulate in VALU (co-executes with XDL WMMA) and complete with
// one shfl_xor(16). Synthetic (MoCHi) negatives are folded analytically.
// Anchor staging to LDS uses the Tensor Data Mover (tensor_load_to_lds).
// ---------------------------------------------------------------------------

typedef __attribute__((ext_vector_type(2))) float        v2f;
typedef __attribute__((ext_vector_type(8))) float        v8f;
typedef __attribute__((ext_vector_type(4))) unsigned int v4u;
typedef __attribute__((ext_vector_type(8))) int          v8i;
typedef __attribute__((ext_vector_type(4))) int          v4i;

#define TAU_INV   10.0f
#define EPS_NORM  1e-12f
#define EPS_DENOM 1e-8f
#define MAX_D     1024

__device__ __forceinline__ float waveReduceSum(float v) {
#pragma unroll
    for (int m = 16; m >= 1; m >>= 1) v += __shfl_xor(v, m, 32);
    return v;
}

// TDM: DMA a 1-D row of D f32 elements from global into LDS (ISA ch.8 D#).
__device__ __forceinline__ void tdm_load_row_to_lds(const float* gsrc,
                                                    unsigned lds_off, int D) {
    const unsigned long long ga = (unsigned long long)(uintptr_t)gsrc;
    v4u g0;
    g0.x = 1u;                                   // count=1, is_restore=0
    g0.y = lds_off;                              // lds_addr [63:32]
    g0.z = (unsigned)(ga & 0xFFFFFFFFull);       // global_addr [95:64]
    g0.w = (unsigned)((ga >> 32) & 0x01FFFFFFull)// global_addr [120:96]
         | (2u << 30);                           // type=2 ("image") [127:126]
    v8i g1;
    g1[0] = (int)(2u << 16);                     // wg_mask=0, data_size=2 (4B)
    g1[1] = (int)(((unsigned)D & 0xFFFFu) << 16);// tensor_dim0[15:0] @ [63:48]
    g1[2] = (int)((((unsigned)D >> 16) & 0xFFFFu)// tensor_dim0[31:16] @ [79:64]
         | (1u << 16));                          // tensor_dim1=1     @ [95:80]
    g1[3] = (int)(((unsigned)D & 0xFFFFu) << 16);// tile_dim0=D       @[127:112]
    g1[4] = (int)(1u | (1u << 16));              // tile_dim1=1, tile_dim2=1
    g1[5] = D;                                   // tensor_dim0_stride[31:0]
    g1[6] = 0;                                   // stride hi, dim1_stride lo
    g1[7] = 0;
    const v4i z4 = {0, 0, 0, 0};                 // groups 2/3: dims 3/4 unused
    const v8i z8 = {0, 0, 0, 0, 0, 0, 0, 0};
    // 6-arg form on this toolchain (clang-23 / therock-10.0 headers).
    __builtin_amdgcn_tensor_load_to_lds(g0, g1, z4, z4, z8, 0);
}

// ---------------- K0: normalize anchor, zero accumulators ------------------
__global__ void k_init(const float* __restrict__ a, float* __restrict__ a_norm,
                       float* __restrict__ S_acc, float* __restrict__ loss_acc,
                       int D) {
    __shared__ float red[32];
    const int tid = threadIdx.x;
    float v  = (tid < D) ? a[tid] : 0.0f;
    float sq = v * v;
    float w  = waveReduceSum(sq);
    const int lane = tid & 31, wid = tid >> 5;
    if (lane == 0) red[wid] = w;
    __syncthreads();
    const int nw = (blockDim.x + 31) >> 5;
    float tot = (tid < nw) ? red[tid] : 0.0f;
    tot = waveReduceSum(tot);
    if (tid == 0) red[0] = tot;
    __syncthreads();
    const float inv = 1.0f / fmaxf(sqrtf(red[0]), EPS_NORM);
    if (tid < D) a_norm[tid] = v * inv;
    if (tid == 0) { *S_acc = 0.0f; *loss_acc = 0.0f; }
}

// ---------------- K1/K3: WMMA GEMV rows . a_norm ---------------------------
// mode 0: accumulate sum(exp(logit)) into S_acc              (negatives pass)
// mode 1: accumulate sum(log(exp(l)+S+eps) - l) into loss_acc (positives)
// DD > 0: compile-time D; DD == 0: runtime D fallback.
template <int DD>
__global__ void k_gemv_wmma(const float* __restrict__ rows, int nrows, int Din,
                            const float* __restrict__ a_norm,
                            float* __restrict__ S_acc,
                            float* __restrict__ loss_acc, int mode) {
    const int D = (DD > 0) ? DD : Din;
    __shared__ float a_lds[MAX_D];

    // Stage the normalized anchor via the Tensor Data Mover (wave 0 only;
    // EXEC is ignored for tensor ops, branch is wave-uniform).
    if (threadIdx.x < 32) {
        tdm_load_row_to_lds(a_norm, (unsigned)(uintptr_t)&a_lds[0], D);
        __builtin_amdgcn_s_wait_tensorcnt(0);
    }
    __syncthreads();

    const int lane    = threadIdx.x & 31;
    const int wpb     = blockDim.x >> 5;
    const int waveId  = blockIdx.x * wpb + (threadIdx.x >> 5);
    const int nWaves  = gridDim.x * wpb;
    const int nTiles  = nrows >> 4;           // 16 rows per tile
    const int halfOff = (lane < 16) ? 0 : 2;  // K phase per wave half
    const int r       = lane & 15;            // row within tile (== B column)

    float S = 0.0f;
    if (mode == 1) S = *S_acc;                // uniform; finalized by K2

    float acc = 0.0f;
    for (int tile = waveId; tile < nTiles; tile += nWaves) {
        const float* rp = rows + (size_t)(tile * 16 + r) * (size_t)D + halfOff;
        v8f   c   = {};
        float nsq = 0.0f;
#pragma unroll 8
        for (int t = 0; t < D; t += 4) {
            // B: 16 h-rows, K-chunk of 4 (global_load_b64 per lane)
            const v2f bv = *(const v2f*)(rp + t);
            // A: anchor chunk replicated across M (LDS broadcast, ds_load_b64)
            const v2f av = *(const v2f*)(&a_lds[t + halfOff]);
            c = __builtin_amdgcn_wmma_f32_16x16x4_f32(
                    false, av, false, bv, (short)0, c, false, false);
            nsq = fmaf(bv.x, bv.x, nsq);      // row-norm partial, co-executes
            nsq = fmaf(bv.y, bv.y, nsq);
        }
        // complete ||row||^2: other wave-half holds the K%4 in {2,3} partial
        const float nsqFull = nsq + __shfl_xor(nsq, 16, 32);
        const float s = c[0];                 // dot(row[r], a_norm), all lanes
        const float d = s / fmaxf(sqrtf(nsqFull), EPS_NORM);
        const float l = d * TAU_INV;
        float term;
        if (mode == 0) term = expf(l);
        else           term = logf(expf(l) + S + EPS_DENOM) - l;
        acc += (lane < 16) ? term : 0.0f;     // wave halves are duplicates
    }
    acc = waveReduceSum(acc);
    if (lane == 0) atomicAdd((mode == 0) ? S_acc : loss_acc, acc);
}

// ---------------- K2: synthetic (MoCHi) negatives, analytic ----------------
__global__ void k_synth(const float* __restrict__ h, int D,
                        const float* __restrict__ a_norm,
                        const int* __restrict__ mix_idx,
                        const int* __restrict__ idx_a,
                        const int* __restrict__ idx_b,
                        const float* __restrict__ alpha_raw,
                        const float* __restrict__ beta_raw,
                        int n_mix, float* __restrict__ S_acc) {
    const int lane = threadIdx.x & 31;
    const int wave = (blockIdx.x * blockDim.x + threadIdx.x) >> 5;
    if (wave >= 2 * n_mix) return;

    float e;
    if (wave < n_mix) {
        // hardest_i = l2norm((1-a)*h_n[mix] + a*a_n)
        const int i = wave;
        const float* rp = h + (size_t)mix_idx[i] * (size_t)D;
        float s = 0.0f, nsq = 0.0f;
        for (int k = lane; k < D; k += 32) {
            const float v = rp[k];
            s   = fmaf(v, a_norm[k], s);
            nsq = fmaf(v, v, nsq);
        }
        s = waveReduceSum(s);  nsq = waveReduceSum(nsq);
        const float d  = s / fmaxf(sqrtf(nsq), EPS_NORM);
        const float al = alpha_raw[i] * 0.4f + 0.1f;
        const float om = 1.0f - al;
        const float num = om * d + al;                 // a_n . a_n == 1
        const float den = fmaxf(sqrtf(om * om + al * al + 2.0f * al * om * d),
                                EPS_NORM);
        e = expf(num / den * TAU_INV);
    } else {
        // harder_i = l2norm(b*h_n[a] + (1-b)*h_n[b])
        const int i = wave - n_mix;
        const float* ra = h + (size_t)idx_a[i] * (size_t)D;
        const float* rb = h + (size_t)idx_b[i] * (size_t)D;
        float sa = 0.0f, sb = 0.0f, na = 0.0f, nb = 0.0f, dab = 0.0f;
        for (int k = lane; k < D; k += 32) {
            const float va = ra[k], vb = rb[k], an = a_norm[k];
            sa  = fmaf(va, an, sa);  sb  = fmaf(vb, an, sb);
            na  = fmaf(va, va, na);  nb  = fmaf(vb, vb, nb);
            dab = fmaf(va, vb, dab);
        }
        sa = waveReduceSum(sa);  sb = waveReduceSum(sb);
        na = waveReduceSum(na);  nb = waveReduceSum(nb);
        dab = waveReduceSum(dab);
        const float ina = 1.0f / fmaxf(sqrtf(na), EPS_NORM);
        const float inb = 1.0f / fmaxf(sqrtf(nb), EPS_NORM);
        const float da = sa * ina, db = sb * inb, cab = dab * ina * inb;
        const float be = beta_raw[i] * 0.4f + 0.3f;
        const float ob = 1.0f - be;
        const float num = be * da + ob * db;
        const float den = fmaxf(sqrtf(be * be + ob * ob + 2.0f * be * ob * cab),
                                EPS_NORM);
        e = expf(num / den * TAU_INV);
    }
    if (lane == 0) atomicAdd(S_acc, e);
}

// ---------------- K4: finalize ---------------------------------------------
__global__ void k_final(const float* __restrict__ loss_acc,
                        float* __restrict__ out, float invN) {
    if (threadIdx.x == 0 && blockIdx.x == 0) out[0] = loss_acc[0] * invN;
}

// ---------------------------------------------------------------------------
extern "C" void kernel_launch(void* const* d_in, const int* in_sizes, int n_in,
                              void* d_out, int out_size, void* d_ws,
                              size_t ws_size, hipStream_t stream) {
    const float* anchor    = (const float*)d_in[0];
    const float* positives = (const float*)d_in[1];
    const float* hardneg   = (const float*)d_in[2];
    const int*   mix_idx   = (const int*)d_in[3];
    const int*   idx_a     = (const int*)d_in[4];
    const int*   idx_b     = (const int*)d_in[5];
    const float* alpha_raw = (const float*)d_in[6];
    const float* beta_raw  = (const float*)d_in[7];

    const int D      = in_sizes[0];      // 512
    const int n_pos  = in_sizes[1] / D;  // 8192
    const int n_hard = in_sizes[2] / D;  // 65536
    const int n_mix  = in_sizes[3];      // 64

    float* ws       = (float*)d_ws;
    float* a_norm   = ws;                // D floats
    float* S_acc    = ws + MAX_D;        // 1 float
    float* loss_acc = ws + MAX_D + 1;    // 1 float

    k_init<<<1, D, 0, stream>>>(anchor, a_norm, S_acc, loss_acc, D);

    const int threads = 256;
    const int wpb     = threads / 32;

    const int tiles_neg = n_hard / 16;                        // 4096
    int blocks_neg = (tiles_neg + wpb - 1) / wpb;             // 512
    if (blocks_neg > 2048) blocks_neg = 2048;

    const int tiles_pos  = n_pos / 16;                        // 512
    const int blocks_pos = (tiles_pos + wpb - 1) / wpb;       // 64

    if (D == 512) {
        k_gemv_wmma<512><<<blocks_neg, threads, 0, stream>>>(
            hardneg, n_hard, D, a_norm, S_acc, loss_acc, /*mode=*/0);
    } else {
        k_gemv_wmma<0><<<blocks_neg, threads, 0, stream>>>(
            hardneg, n_hard, D, a_norm, S_acc, loss_acc, /*mode=*/0);
    }

    const int synthThreads = 2 * n_mix * 32;                  // 4096 lanes
    k_synth<<<(synthThreads + threads - 1) / threads, threads, 0, stream>>>(
        hardneg, D, a_norm, mix_idx, idx_a, idx_b, alpha_raw, beta_raw,
        n_mix, S_acc);

    if (D == 512) {
        k_gemv_wmma<512><<<blocks_pos, threads, 0, stream>>>(
            positives, n_pos, D, a_norm, S_acc, loss_acc, /*mode=*/1);
    } else {
        k_gemv_wmma<0><<<blocks_pos, threads, 0, stream>>>(
            positives, n_pos, D, a_norm, S_acc, loss_acc, /*mode=*/1);
    }

    k_final<<<1, 1, 0, stream>>>(loss_acc, (float*)d_out, 1.0f / (float)n_pos);
}